// HighOrderActivationA_16741782520152
// MI455X (gfx1250) — compile-verified
//
#include <hip/hip_runtime.h>

// HighOrderActivation on MI455X (gfx1250, wave32).
//
// out[b, g*16+o] = sum_k coef[b,g,k] * params[g, idx[b,g,k], o]
// == per (16-row b-tile, g):  OutTile(16x16) = S(16x16) @ P_g(16x16)
// where S[row, t] = sum_k coef_k * (idx_k == t)  (4 distinct nonzeros/row).
// Chained as 4x V_WMMA_F32_16X16X4_F32 (K=4 slices).
//
// Memory-bound problem (~160 MB total vs 23.3 TB/s => ~7us floor); params
// (256 KB) stays resident in the 192 MB L2, X staged via LDS for coalescing.

typedef __attribute__((ext_vector_type(2))) float v2f;
typedef __attribute__((ext_vector_type(8))) float v8f;

#define G_CNT   256
#define ARITY_  4
#define O_CNT   16
#define TBL     16          // 2^ARITY
#define GPB     8           // groups per block = waves per block
#define XS_STRIDE 36        // 16 rows map to 16 distinct LDS banks (r*36 mod 64 all distinct)

// Branchless compare-exchange carrying (value, original-index) pairs.
// Ties never swap (strict >); tied elements contribute zero coef anyway.
#define CSWAP(va, vb, ia, ib)                                   \
  {                                                             \
    bool sw_ = (va > vb);                                       \
    float t0_ = sw_ ? vb : va;                                  \
    float t1_ = sw_ ? va : vb;                                  \
    int   j0_ = sw_ ? ib : ia;                                  \
    int   j1_ = sw_ ? ia : ib;                                  \
    va = t0_; vb = t1_; ia = j0_; ib = j1_;                     \
  }

__global__ __launch_bounds__(256) void hoact_wmma_kernel(
    const float* __restrict__ X,      // [B, G*4]
    const float* __restrict__ P,      // [G, 16, 16]
    float* __restrict__ out)          // [B, G*16]
{
  __shared__ float Xs[16 * XS_STRIDE];

  const int tid  = threadIdx.x;
  const int wave = tid >> 5;
  const int lane = tid & 31;
  const int hi   = lane >> 4;     // half-wave: 0 or 1
  const int col  = lane & 15;     // N column for B/C, also this lane's S-row
  const int tile = blockIdx.y;    // 16-row tile of B
  const int g_base = blockIdx.x * GPB;
  const int g      = g_base + wave;

  // ---- Stage X tile: 16 rows x (GPB*ARITY = 32) floats, coalesced (128B/row)
  const long long rowBase =
      (long long)(tile * 16) * (G_CNT * ARITY_) + (long long)g_base * ARITY_;
  #pragma unroll
  for (int i = tid; i < 16 * (GPB * ARITY_); i += 256) {
    const int r = i >> 5;
    const int c = i & 31;
    Xs[r * XS_STRIDE + c] = X[rowBase + (long long)r * (G_CNT * ARITY_) + c];
  }
  __syncthreads();

  // ---- Per lane: both half-waves redundantly process row = col of the tile,
  // so every lane owns its row's coef/idx and no cross-lane shuffle is needed
  // to build the WMMA A-operand layout.
  const float* xr = &Xs[col * XS_STRIDE + wave * ARITY_];
  float v0 = xr[0], v1 = xr[1], v2 = xr[2], v3 = xr[3];
  int i0 = 0, i1 = 1, i2 = 2, i3 = 3;

  // 5-compare sorting network, ascending.
  CSWAP(v0, v1, i0, i1);
  CSWAP(v2, v3, i2, i3);
  CSWAP(v0, v2, i0, i2);
  CSWAP(v1, v3, i1, i3);
  CSWAP(v1, v2, i1, i2);

  // Barycentric coefficients and suffix-sum bitmask indices.
  const float c0 = v0, c1 = v1 - v0, c2 = v2 - v1, c3 = v3 - v2;
  const int p0 = 1 << i0, p1 = 1 << i1, p2 = 1 << i2, p3 = 1 << i3;
  const int idx3 = p3;
  const int idx2 = p3 + p2;
  const int idx1 = idx2 + p1;
  const int idx0 = idx1 + p0;   // always 15

  const float* __restrict__ Pg = P + (long long)g * (TBL * O_CNT);

  // ---- OutTile = S @ P_g via 4 chained V_WMMA_F32_16X16X4_F32.
  // A (16x4 f32, 2 VGPRs): lane L holds M=L&15; VGPR0/1 hold local K = 2*hi + {0,1}.
  // B (4x16 f32, 2 VGPRs): lane L holds N=L&15; VGPR0/1 hold local K = 2*hi + {0,1}.
  v8f acc = {};
  #pragma unroll
  for (int ck = 0; ck < 4; ++ck) {
    const int K0 = ck * 4 + hi * 2;     // global K of this lane's first element
    const int K1 = K0 + 1;

    v2f a;
    a.x = (idx0 == K0 ? c0 : 0.0f) + (idx1 == K0 ? c1 : 0.0f) +
          (idx2 == K0 ? c2 : 0.0f) + (idx3 == K0 ? c3 : 0.0f);
    a.y = (idx0 == K1 ? c0 : 0.0f) + (idx1 == K1 ? c1 : 0.0f) +
          (idx2 == K1 ? c2 : 0.0f) + (idx3 == K1 ? c3 : 0.0f);

    v2f b;
    b.x = Pg[K0 * O_CNT + col];
    b.y = Pg[K1 * O_CNT + col];

    // 8 args: (neg_a, A, neg_b, B, c_mod, C, reuse_a, reuse_b)
    acc = __builtin_amdgcn_wmma_f32_16x16x4_f32(
        false, a, false, b, (short)0, acc, false, false);
  }

  // ---- Store C: VGPR r -> (M = r + 8*hi, N = col)
  const long long outBase =
      (long long)(tile * 16) * (G_CNT * O_CNT) + (long long)g * O_CNT + col;
  #pragma unroll
  for (int r = 0; r < 8; ++r) {
    const int M = r + hi * 8;
    out[outBase + (long long)M * (G_CNT * O_CNT)] = acc[r];
  }
}

extern "C" void kernel_launch(void* const* d_in, const int* in_sizes, int n_in,
                              void* d_out, int out_size, void* d_ws, size_t ws_size,
                              hipStream_t stream) {
  (void)n_in; (void)out_size; (void)d_ws; (void)ws_size;
  const float* X = (const float*)d_in[0];   // [B, 1024] f32
  const float* P = (const float*)d_in[1];   // [256, 16, 16] f32
  float* out = (float*)d_out;               // [B, 4096] f32

  const int Btot = in_sizes[0] / (G_CNT * ARITY_);   // 8192
  dim3 grid(G_CNT / GPB, Btot / 16);                 // (32, 512)
  dim3 block(256);                                   // 8 wave32 waves
  hipLaunchKernelGGL(hoact_wmma_kernel, grid, block, 0, stream, X, P, out);
}